// VectorQuantizer_16501264352054
// MI455X (gfx1250) — compile-verified
//
#include <hip/hip_runtime.h>

typedef float v2f __attribute__((ext_vector_type(2)));
typedef float v8f __attribute__((ext_vector_type(8)));

#define K_ENTRIES 8192
#define C_CH 8
#define HW 4096                      // 64*64
#define N_POINTS 32768               // 8*64*64
#define NTILES (N_POINTS / 16)       // 2048
#define WAVES_PER_BLOCK 8
#define NBLOCKS (NTILES / WAVES_PER_BLOCK)   // 256
#define CB_FLOATS (K_ENTRIES * C_CH)         // 65536

// Main kernel: one 16-point tile per wave32; full codebook resident in LDS.
__global__ __launch_bounds__(256) void vq_argmin_kernel(
    const float* __restrict__ z, const float* __restrict__ cb,
    float* __restrict__ out, float* __restrict__ wsum)
{
    extern __shared__ float smem[];
    float* s_cb = smem;              // codebook scaled by -2: [8192][8]
    float* s_c2 = smem + CB_FLOATS;  // ||c_k||^2: [8192]

    const int tid = threadIdx.x;

    // ---- Stage codebook into LDS (scaled by -2) + row norms ----
    for (int r = tid; r < K_ENTRIES; r += 256) {
        float4 a = ((const float4*)cb)[r * 2 + 0];
        float4 b = ((const float4*)cb)[r * 2 + 1];
        float c2 = a.x*a.x + a.y*a.y + a.z*a.z + a.w*a.w
                 + b.x*b.x + b.y*b.y + b.z*b.z + b.w*b.w;
        ((float4*)s_cb)[r * 2 + 0] = make_float4(-2.f*a.x, -2.f*a.y, -2.f*a.z, -2.f*a.w);
        ((float4*)s_cb)[r * 2 + 1] = make_float4(-2.f*b.x, -2.f*b.y, -2.f*b.z, -2.f*b.w);
        s_c2[r] = c2;
    }
    __syncthreads();

    const int wave = tid >> 5;
    const int lane = tid & 31;
    const int lh   = lane >> 4;      // lane half: 0 or 1
    const int m    = lane & 15;      // tile-local point / column id
    const int tile = blockIdx.x * WAVES_PER_BLOCK + wave;
    const int p    = tile * 16 + m;  // global point id

    // z layout [B, C, H, W]: channel stride = HW, batch stride = C*HW
    const int zbase = (p >> 12) * (C_CH * HW) + (p & (HW - 1));
    const int c0 = lh * 2;           // this lane's channel group: {c0,c0+1,c0+4,c0+5}
    const float az0 = z[zbase + (c0    ) * HW];
    const float az1 = z[zbase + (c0 + 1) * HW];
    const float az2 = z[zbase + (c0 + 4) * HW];
    const float az3 = z[zbase + (c0 + 5) * HW];
    // A fragments (16x4 f32): VGPR0/1 = K{0,1} for lanes 0-15, K{2,3} for 16-31
    const v2f a_lo = { az0, az1 };   // channels 0..3 across the wave
    const v2f a_hi = { az2, az3 };   // channels 4..7

    float bv[8];
    int   bi[8];
    #pragma unroll
    for (int r = 0; r < 8; ++r) { bv[r] = __builtin_inff(); bi[r] = 0; }

    const float* cbp = s_cb + (size_t)m * C_CH + c0;  // this lane's B slice base
    const float* c2p = s_c2 + m;

    // ---- Sweep all 8192 codebook entries, 16 columns per step ----
    #pragma unroll 4
    for (int kb = 0; kb < K_ENTRIES; kb += 16) {
        const int col = kb + m;
        // B fragments (4x16 f32): lower lanes rows K{0,1}, upper rows K{2,3}
        const v2f b0 = *(const v2f*)(cbp + (size_t)kb * C_CH);       // ch c0,c0+1
        const v2f b1 = *(const v2f*)(cbp + (size_t)kb * C_CH + 4);   // ch c0+4,c0+5
        const float c2 = c2p[kb];
        v8f acc = { c2, c2, c2, c2, c2, c2, c2, c2 };  // C init = ||c_col||^2
        // score = ||c||^2 - 2 z.c   (codebook pre-scaled by -2)
        acc = __builtin_amdgcn_wmma_f32_16x16x4_f32(false, a_lo, false, b0,
                                                    (short)0, acc, false, false);
        acc = __builtin_amdgcn_wmma_f32_16x16x4_f32(false, a_hi, false, b1,
                                                    (short)0, acc, false, false);
        #pragma unroll
        for (int r = 0; r < 8; ++r) {
            const float v = acc[r];
            const bool lt = v < bv[r];          // strict: keeps lowest k on ties
            bv[r] = lt ? v   : bv[r];
            bi[r] = lt ? col : bi[r];
        }
    }

    // ---- Argmin butterfly within each 16-lane half (tie-break: lowest k) ----
    #pragma unroll
    for (int mask = 1; mask <= 8; mask <<= 1) {
        #pragma unroll
        for (int r = 0; r < 8; ++r) {
            const float ov = __shfl_xor(bv[r], mask, 32);
            const int   oi = __shfl_xor(bi[r], mask, 32);
            const bool take = (ov < bv[r]) || (ov == bv[r] && oi < bi[r]);
            bv[r] = take ? ov : bv[r];
            bi[r] = take ? oi : bi[r];
        }
    }
    // Route winning index of point m to its owner lane (rows 0-7 live in lower
    // half lanes, rows 8-15 in upper half lanes).
    int myk = 0;
    #pragma unroll
    for (int r = 0; r < 8; ++r) {
        const int own  = bi[r];
        const int oth  = __shfl_xor(own, 16, 32);
        const int cand = ((m >> 3) == lh) ? own : oth;
        if ((m & 7) == r) myk = cand;
    }

    // ---- Emit z_q (unscale -2x codebook by -0.5: exact in fp32) ----
    const v2f q01 = *(const v2f*)(s_cb + (size_t)myk * C_CH + c0);
    const v2f q23 = *(const v2f*)(s_cb + (size_t)myk * C_CH + 4 + c0);
    const float zq0 = -0.5f * q01.x, zq1 = -0.5f * q01.y;
    const float zq2 = -0.5f * q23.x, zq3 = -0.5f * q23.y;
    out[zbase + (c0    ) * HW] = zq0;
    out[zbase + (c0 + 1) * HW] = zq1;
    out[zbase + (c0 + 4) * HW] = zq2;
    out[zbase + (c0 + 5) * HW] = zq3;

    // ---- Per-wave loss partial: sum (zq - z)^2 ----
    float dl = (zq0 - az0) * (zq0 - az0) + (zq1 - az1) * (zq1 - az1)
             + (zq2 - az2) * (zq2 - az2) + (zq3 - az3) * (zq3 - az3);
    #pragma unroll
    for (int mask = 1; mask < 32; mask <<= 1) dl += __shfl_xor(dl, mask, 32);
    if (lane == 0) wsum[tile] = dl;
}

// Deterministic tree reduction of the 2048 wave partials -> loss scalar.
__global__ __launch_bounds__(256) void vq_loss_reduce(
    const float* __restrict__ wsum, float* __restrict__ out)
{
    __shared__ float s[256];
    const int t = threadIdx.x;
    float acc = 0.f;
    for (int i = t; i < NTILES; i += 256) acc += wsum[i];
    s[t] = acc;
    __syncthreads();
    for (int off = 128; off > 0; off >>= 1) {
        if (t < off) s[t] += s[t + off];
        __syncthreads();
    }
    // loss = (beta + 1) * mean = 1.25 * sum / 262144
    if (t == 0) out[N_POINTS * C_CH] = s[0] * (1.25f / (float)(N_POINTS * C_CH));
}

extern "C" void kernel_launch(void* const* d_in, const int* in_sizes, int n_in,
                              void* d_out, int out_size, void* d_ws, size_t ws_size,
                              hipStream_t stream) {
    const float* z  = (const float*)d_in[0];   // [8,8,64,64]
    const float* cb = (const float*)d_in[1];   // [8192,8]
    float* out = (float*)d_out;                // [262144] z_q  + [1] loss
    float* ws  = (float*)d_ws;                 // >= 2048 floats

    const size_t smem = (size_t)(CB_FLOATS + K_ENTRIES) * sizeof(float); // 288 KB
    vq_argmin_kernel<<<NBLOCKS, 256, smem, stream>>>(z, cb, out, ws);
    vq_loss_reduce<<<1, 256, 0, stream>>>(ws, out);
}